// Extractor_71021579207093
// MI455X (gfx1250) — compile-verified
//
#include <hip/hip_runtime.h>
#include <math.h>

typedef float v2f __attribute__((ext_vector_type(2)));
typedef float v8f __attribute__((ext_vector_type(8)));

#define S_PTS 9
#define N_HALF 4

// Output layout (floats, concatenated in reference return order), NP = b*h*w:
//   fusion_values : [0,        9*NP)
//   fusion_weights: [9*NP,    18*NP)
//   ray_pts       : [18*NP,   45*NP)
//   depth_flat    : [45*NP,   46*NP)
//   indices       : [46*NP,  262*NP)
//   interp_weights: [262*NP, 334*NP)
//   coords        : [334*NP, 337*NP)

__global__ __launch_bounds__(256)
void extractor_kernel(const float* __restrict__ depth,
                      const float* __restrict__ extr,
                      const float* __restrict__ intr,
                      const float* __restrict__ vol,
                      const float* __restrict__ origin,
                      const float* __restrict__ resol,
                      const float* __restrict__ wvol,
                      float* __restrict__ out,
                      int npix, int HW, int W, int VD)
{
    const int g    = blockIdx.x * blockDim.x + threadIdx.x;
    const int lane = threadIdx.x & 31;
    const bool inb = (g < npix);
    const int gs   = inb ? g : (npix - 1);      // clamped index: keeps WMMA lanes fed
    const int b    = gs / HW;
    const int pix  = gs - b * HW;
    const int row  = pix / W;                   // xx (row index)
    const int col  = pix - row * W;             // yy (col index)

    // ---------- per-batch combined transform T = E * [[K^-1,0],[0,1]] ----------
    const float* E = extr + b * 16;
    const float* K = intr + b * 9;
    const float a00=K[0], a01=K[1], a02=K[2];
    const float a10=K[3], a11=K[4], a12=K[5];
    const float a20=K[6], a21=K[7], a22=K[8];
    const float det = a00*(a11*a22 - a12*a21) - a01*(a10*a22 - a12*a20) + a02*(a10*a21 - a11*a20);
    const float inv = 1.0f / det;
    const float i00=(a11*a22-a12*a21)*inv, i01=(a02*a21-a01*a22)*inv, i02=(a01*a12-a02*a11)*inv;
    const float i10=(a12*a20-a10*a22)*inv, i11=(a00*a22-a02*a20)*inv, i12=(a02*a10-a00*a12)*inv;
    const float i20=(a10*a21-a11*a20)*inv, i21=(a01*a20-a00*a21)*inv, i22=(a00*a11-a01*a10)*inv;

    float4 T0, T1, T2, T3;
    {
        float e0=E[0], e1=E[1], e2=E[2], e3=E[3];
        T0 = make_float4(e0*i00+e1*i10+e2*i20, e0*i01+e1*i11+e2*i21, e0*i02+e1*i12+e2*i22, e3);
        e0=E[4]; e1=E[5]; e2=E[6]; e3=E[7];
        T1 = make_float4(e0*i00+e1*i10+e2*i20, e0*i01+e1*i11+e2*i21, e0*i02+e1*i12+e2*i22, e3);
        e0=E[8]; e1=E[9]; e2=E[10]; e3=E[11];
        T2 = make_float4(e0*i00+e1*i10+e2*i20, e0*i01+e1*i11+e2*i21, e0*i02+e1*i12+e2*i22, e3);
        e0=E[12]; e1=E[13]; e2=E[14]; e3=E[15];
        T3 = make_float4(e0*i00+e1*i10+e2*i20, e0*i01+e1*i11+e2*i21, e0*i02+e1*i12+e2*i22, e3);
    }

    const float zval = depth[gs];
    const float u = (float)col * zval;   // u = yy * z
    const float v = (float)row * zval;   // v = xx * z

    // ---------- world coords via V_WMMA_F32_16X16X4_F32 (wave32, EXEC all-1s) ---
    // A (16x4): row m = T row m (m<4), else 0.  32b A layout: lanes0-15 M=0-15,
    //   VGPR0 = K0 | K2(lanes16-31), VGPR1 = K1 | K3.
    // B (4x16): column n = [u_n, v_n, z_n, 1].  32b B layout: lanes0-15 hold
    //   K=0,1 (V0,V1), lanes16-31 hold K=2,3.
    // D (16x16): row m in VGPR m, lanes0-15 = N=0..15  ->  rows 0..2 = x,y,z.
    float cxw, cyw, czw;
    {
        const int  m  = lane & 15;
        const bool lo = (lane < 16);
        const float t0a = lo ? T0.x : T0.z, t0b = lo ? T0.y : T0.w;
        const float t1a = lo ? T1.x : T1.z, t1b = lo ? T1.y : T1.w;
        const float t2a = lo ? T2.x : T2.z, t2b = lo ? T2.y : T2.w;
        const float t3a = lo ? T3.x : T3.z, t3b = lo ? T3.y : T3.w;
        v2f A;
        A.x = (m==0)?t0a:(m==1)?t1a:(m==2)?t2a:(m==3)?t3a:0.0f;
        A.y = (m==0)?t0b:(m==1)?t1b:(m==2)?t2b:(m==3)?t3b:0.0f;

        const int n1 = lane & 15, n2 = n1 + 16;
        const float u1=__shfl(u,n1,32), v1=__shfl(v,n1,32), z1=__shfl(zval,n1,32);
        const float u2=__shfl(u,n2,32), v2=__shfl(v,n2,32), z2=__shfl(zval,n2,32);
        v2f B1, B2;
        B1.x = lo ? u1 : z1;  B1.y = lo ? v1 : 1.0f;   // K0/K1 | K2/K3
        B2.x = lo ? u2 : z2;  B2.y = lo ? v2 : 1.0f;

        v8f c0 = {};
        v8f D1, D2;
#if defined(__HIP_DEVICE_COMPILE__)
        D1 = __builtin_amdgcn_wmma_f32_16x16x4_f32(false, A, false, B1, (short)0, c0, false, false);
        D2 = __builtin_amdgcn_wmma_f32_16x16x4_f32(false, A, false, B2, (short)0, c0, false, false);
#else
        D1 = c0; D2 = c0;
#endif
        const float x2 = __shfl(D2[0], n1, 32);
        const float y2 = __shfl(D2[1], n1, 32);
        const float zz2= __shfl(D2[2], n1, 32);
        cxw = lo ? D1[0] : x2;
        cyw = lo ? D1[1] : y2;
        czw = lo ? D1[2] : zz2;
    }

    if (!inb) return;   // after WMMA: tail lanes drop out only now

    // Single reciprocal instead of 6 IEEE divisions for the /res scaling.
    const float rres = 1.0f / resol[0];
    const float ox = origin[0], oy = origin[1], oz = origin[2];

    // voxel-space center and eye, ray direction
    const float cvx = (cxw - ox) * rres;
    const float cvy = (cyw - oy) * rres;
    const float cvz = (czw - oz) * rres;
    const float evx = (E[3]  - ox) * rres;
    const float evy = (E[7]  - oy) * rres;
    const float evz = (E[11] - oz) * rres;
    float dx = cvx - evx, dy = cvy - evy, dz = cvz - evz;
    const float nrm = sqrtf(dx*dx + dy*dy + dz*dz);
    const float inv_dn = 1.0f / fmaxf(nrm, 1e-12f);   // one division for normalize
    dx *= inv_dn; dy *= inv_dn; dz *= inv_dn;

    const long long NP = npix;
    float* __restrict__ out_fv = out;
    float* __restrict__ out_fw = out +   9LL*NP;
    float* __restrict__ out_rp = out +  18LL*NP;
    float* __restrict__ out_dz = out +  45LL*NP;
    float* __restrict__ out_id = out +  46LL*NP;
    float* __restrict__ out_iw = out + 262LL*NP;
    float* __restrict__ out_co = out + 334LL*NP;

    out_dz[g] = zval;
    out_co[(long long)g*3 + 0] = cxw;
    out_co[(long long)g*3 + 1] = cyw;
    out_co[(long long)g*3 + 2] = czw;

    const float VDf = (float)VD;

    #pragma unroll
    for (int s = 0; s < S_PTS; ++s) {
        const float off = (float)(s - N_HALF);
        const float px = fmaf(off, dx, cvx);
        const float py = fmaf(off, dy, cvy);
        const float pz = fmaf(off, dz, cvz);

        const long long base = (long long)g * S_PTS + s;
        out_rp[base*3 + 0] = px;
        out_rp[base*3 + 1] = py;
        out_rp[base*3 + 2] = pz;

        const float flx = floorf(px), fly = floorf(py), flz = floorf(pz);
        const float dfx = fabsf(px - (flx + 0.5f));
        const float dfy = fabsf(py - (fly + 0.5f));
        const float dfz = fabsf(pz - (flz + 0.5f));

        const float wx[2] = {1.0f - dfx, dfx};
        const float wy[2] = {1.0f - dfy, dfy};
        const float wz[2] = {1.0f - dfz, dfz};

        const float fx1 = flx + 1.0f, fy1 = fly + 1.0f, fz1 = flz + 1.0f;
        const int ix[2] = { (int)fminf(fmaxf(flx,0.0f), VDf-1.0f),
                            (int)fminf(fmaxf(fx1,0.0f), VDf-1.0f) };
        const int iy[2] = { (int)fminf(fmaxf(fly,0.0f), VDf-1.0f),
                            (int)fminf(fmaxf(fy1,0.0f), VDf-1.0f) };
        const int iz[2] = { (int)fminf(fmaxf(flz,0.0f), VDf-1.0f),
                            (int)fminf(fmaxf(fz1,0.0f), VDf-1.0f) };
        const float vx[2] = { (flx >= 0.0f && flx < VDf) ? 1.0f : 0.0f,
                              (fx1 >= 0.0f && fx1 < VDf) ? 1.0f : 0.0f };
        const float vy[2] = { (fly >= 0.0f && fly < VDf) ? 1.0f : 0.0f,
                              (fy1 >= 0.0f && fy1 < VDf) ? 1.0f : 0.0f };
        const float vz[2] = { (flz >= 0.0f && flz < VDf) ? 1.0f : 0.0f,
                              (fz1 >= 0.0f && fz1 < VDf) ? 1.0f : 0.0f };

        float sumV = 0.0f, sumW = 0.0f;
        float wts8[8];
        #pragma unroll
        for (int c = 0; c < 8; ++c) {
            const int i = (c >> 2) & 1, j = (c >> 1) & 1, k = c & 1;
            const float wt = wx[i] * wy[j] * wz[k];
            wts8[c] = wt;
            const float valid = vx[i] * vy[j] * vz[k];
            const long long lin = ((long long)ix[i] * VD + iy[j]) * VD + iz[k];
            sumV = fmaf(wt, valid * vol[lin],  sumV);
            sumW = fmaf(wt, valid * wvol[lin], sumW);
        }
        out_fv[base] = sumV;
        out_fw[base] = sumW;

        // indices: 24 contiguous floats per (g,s), 96B aligned -> 6x float4
        float4* idq = (float4*)(out_id + base * 24);
        idq[0] = make_float4(flx, fly, flz, flx);
        idq[1] = make_float4(fly, fz1, flx, fy1);
        idq[2] = make_float4(flz, flx, fy1, fz1);
        idq[3] = make_float4(fx1, fly, flz, fx1);
        idq[4] = make_float4(fly, fz1, fx1, fy1);
        idq[5] = make_float4(flz, fx1, fy1, fz1);

        // interp weights: 8 contiguous floats per (g,s) -> 2x float4
        float4* iwq = (float4*)(out_iw + base * 8);
        iwq[0] = make_float4(wts8[0], wts8[1], wts8[2], wts8[3]);
        iwq[1] = make_float4(wts8[4], wts8[5], wts8[6], wts8[7]);
    }
}

extern "C" void kernel_launch(void* const* d_in, const int* in_sizes, int n_in,
                              void* d_out, int out_size, void* d_ws, size_t ws_size,
                              hipStream_t stream) {
    const float* depth = (const float*)d_in[0];
    const float* extr  = (const float*)d_in[1];
    const float* intr  = (const float*)d_in[2];
    const float* vol   = (const float*)d_in[3];
    const float* orig  = (const float*)d_in[4];
    const float* resol = (const float*)d_in[5];
    const float* wvol  = (const float*)d_in[6];
    float* out = (float*)d_out;

    const int npix = in_sizes[0];          // b*h*w = 153600
    const int B    = in_sizes[1] / 16;     // extrinsics b*4*4
    const int HW   = npix / B;
    const int W    = 320;                  // reference image width (fixed)

    // integer cube root of volume element count -> VD (256)
    int VD = 1;
    const long long nvol = (long long)in_sizes[3];
    while ((long long)(VD + 1) * (VD + 1) * (VD + 1) <= nvol) ++VD;

    const int blocks = (npix + 255) / 256;
    extractor_kernel<<<blocks, 256, 0, stream>>>(depth, extr, intr, vol, orig,
                                                 resol, wvol, out, npix, HW, W, VD);
}